// Corr_46651934769681
// MI455X (gfx1250) — compile-verified
//
#include <hip/hip_runtime.h>
#include <float.h>
#include <math.h>

// Problem constants (B=2, C=256, H=W=60)
#define Hh   60
#define Ww   60
#define HW   3600          // 60*60
#define Cc   256
#define Bb   2
#define KTOP 10

typedef float v2f __attribute__((ext_vector_type(2)));
typedef float v8f __attribute__((ext_vector_type(8)));

// ---------------- output layout (floats) ----------------
#define OFF_X2CAT ((size_t)Bb * 512 * HW)        // 3,686,400
#define OFF_VAL1  ((size_t)2 * Bb * 512 * HW)    // 7,372,800
#define OFF_VAL2  (OFF_VAL1 + (size_t)Bb * KTOP * HW)

// =====================================================================
// Kernel 1: L[b,n,m] = alpha * sum_c x1[b,c,n]*x2[b,c,m]
// f32 WMMA 16x16x4, one wave computes a 16(M) x 48(N) tile, K-loop by 4.
// A-matrix f32 16x4 lane layout: lanes 0-15 -> K=0 (v0), K=1 (v1);
// lanes 16-31 -> K=2 (v0), K=3 (v1). B symmetric with N across lanes.
// =====================================================================
__global__ __launch_bounds__(32) void gemm_wmma_f32(
    const float* __restrict__ x1, const float* __restrict__ x2,
    const float* __restrict__ alpha_p, float* __restrict__ L)
{
    const int lane = threadIdx.x;
    const int half = lane >> 4;     // 0 or 1
    const int l15  = lane & 15;
    const int m0 = blockIdx.x * 48;
    const int n0 = blockIdx.y * 16;
    const int b  = blockIdx.z;
    const float alpha = alpha_p[0];

    const float* A  = x1 + (size_t)b * Cc * HW;
    const float* Bm = x2 + (size_t)b * Cc * HW;

    v8f acc0 = {}; v8f acc1 = {}; v8f acc2 = {};
    for (int c0 = 0; c0 < Cc; c0 += 4) {
        const int ca = c0 + 2 * half;
        const float* ar0 = A  + (size_t)ca * HW;
        const float* ar1 = ar0 + HW;
        const float* br0 = Bm + (size_t)ca * HW;
        const float* br1 = br0 + HW;
        v2f av; av[0] = ar0[n0 + l15];        av[1] = ar1[n0 + l15];
        v2f b0; b0[0] = br0[m0 + l15];        b0[1] = br1[m0 + l15];
        v2f b1; b1[0] = br0[m0 + 16 + l15];   b1[1] = br1[m0 + 16 + l15];
        v2f b2; b2[0] = br0[m0 + 32 + l15];   b2[1] = br1[m0 + 32 + l15];
        acc0 = __builtin_amdgcn_wmma_f32_16x16x4_f32(false, av, false, b0, (short)0, acc0, false, false);
        acc1 = __builtin_amdgcn_wmma_f32_16x16x4_f32(false, av, false, b1, (short)0, acc1, false, false);
        acc2 = __builtin_amdgcn_wmma_f32_16x16x4_f32(false, av, false, b2, (short)0, acc2, false, false);
    }
    // C/D layout: VGPR v -> row = v + 8*half, col = lane&15 (per 16-col subtile)
    float* Lb = L + (size_t)b * HW * HW;
    for (int v = 0; v < 8; ++v) {
        const int row = n0 + v + 8 * half;
        const size_t base = (size_t)row * HW;
        Lb[base + m0 +      l15] = alpha * acc0[v];
        Lb[base + m0 + 16 + l15] = alpha * acc1[v];
        Lb[base + m0 + 32 + l15] = alpha * acc2[v];
    }
}

// =====================================================================
// Kernel 2: row softmax stats (max, sum exp) — one 256-thr WG per row
// =====================================================================
__global__ __launch_bounds__(256) void row_stats(
    const float* __restrict__ L, float* __restrict__ rowmax, float* __restrict__ rowsum)
{
    __shared__ float row[HW];
    __shared__ float red[256];
    const int n = blockIdx.x, b = blockIdx.y, t = threadIdx.x;
    const float* Lr = L + ((size_t)b * HW + n) * HW;
    for (int e = t; e < HW; e += 256) row[e] = Lr[e];
    __syncthreads();
    float lv = -FLT_MAX;
    for (int e = t; e < HW; e += 256) lv = fmaxf(lv, row[e]);
    red[t] = lv; __syncthreads();
    for (int s = 128; s > 0; s >>= 1) { if (t < s) red[t] = fmaxf(red[t], red[t + s]); __syncthreads(); }
    const float mx = red[0]; __syncthreads();
    float ls = 0.f;
    for (int e = t; e < HW; e += 256) ls += expf(row[e] - mx);
    red[t] = ls; __syncthreads();
    for (int s = 128; s > 0; s >>= 1) { if (t < s) red[t] += red[t + s]; __syncthreads(); }
    if (t == 0) { rowmax[b * HW + n] = mx; rowsum[b * HW + n] = red[0]; }
}

// =====================================================================
// Kernels 3-6: column softmax stats, deterministic two-level reduction
// =====================================================================
#define NCHUNK 10
#define ROWS_PER 360

__global__ __launch_bounds__(256) void col_max_part(
    const float* __restrict__ L, float* __restrict__ part)
{
    const int m = blockIdx.x * 256 + threadIdx.x;
    if (m >= HW) return;
    const int chunk = blockIdx.y, b = blockIdx.z;
    const float* Lb = L + (size_t)b * HW * HW;
    float lv = -FLT_MAX;
    for (int n = chunk * ROWS_PER; n < chunk * ROWS_PER + ROWS_PER; ++n)
        lv = fmaxf(lv, Lb[(size_t)n * HW + m]);
    part[((size_t)b * NCHUNK + chunk) * HW + m] = lv;
}

__global__ __launch_bounds__(256) void col_max_comb(
    const float* __restrict__ part, float* __restrict__ colmax)
{
    const int idx = blockIdx.x * 256 + threadIdx.x;
    if (idx >= Bb * HW) return;
    const int m = idx % HW, b = idx / HW;
    float lv = -FLT_MAX;
    for (int c = 0; c < NCHUNK; ++c) lv = fmaxf(lv, part[((size_t)b * NCHUNK + c) * HW + m]);
    colmax[idx] = lv;
}

__global__ __launch_bounds__(256) void col_sum_part(
    const float* __restrict__ L, const float* __restrict__ colmax, float* __restrict__ part)
{
    const int m = blockIdx.x * 256 + threadIdx.x;
    if (m >= HW) return;
    const int chunk = blockIdx.y, b = blockIdx.z;
    const float* Lb = L + (size_t)b * HW * HW;
    const float cm = colmax[b * HW + m];
    float ls = 0.f;
    for (int n = chunk * ROWS_PER; n < chunk * ROWS_PER + ROWS_PER; ++n)
        ls += expf(Lb[(size_t)n * HW + m] - cm);
    part[((size_t)b * NCHUNK + chunk) * HW + m] = ls;
}

__global__ __launch_bounds__(256) void col_sum_comb(
    const float* __restrict__ part, float* __restrict__ colsum)
{
    const int idx = blockIdx.x * 256 + threadIdx.x;
    if (idx >= Bb * HW) return;
    const int m = idx % HW, b = idx / HW;
    float ls = 0.f;
    for (int c = 0; c < NCHUNK; ++c) ls += part[((size_t)b * NCHUNK + c) * HW + m];
    colsum[idx] = ls;
}

// =====================================================================
// Kernel 7: W[n,m] = softmax_row * softmax_col * zero_window mask (in place)
//   = exp(2L - rmax_n - cmax_m)/(rsum_n*csum_m) * (1 - exp(-(dr^2+dc^2)/162))
// =====================================================================
__global__ __launch_bounds__(256) void compute_w(
    float* __restrict__ L, const float* __restrict__ rowmax, const float* __restrict__ rowsum,
    const float* __restrict__ colmax, const float* __restrict__ colsum)
{
    const size_t idx = (size_t)blockIdx.x * 256 + threadIdx.x;
    if (idx >= (size_t)Bb * HW * HW) return;
    const int m = (int)(idx % HW);
    const size_t t2 = idx / HW;
    const int n = (int)(t2 % HW);
    const int b = (int)(t2 / HW);
    const float l  = L[idx];
    const float sm = expf(2.f * l - rowmax[b * HW + n] - colmax[b * HW + m])
                   / (rowsum[b * HW + n] * colsum[b * HW + m]);
    const int nr = n / 60, nc = n - nr * 60;
    const int mr = m / 60, mc = m - mr * 60;
    const float dr = (float)(mr - nr), dc = (float)(mc - nc);
    const float g  = expf(-(dr * dr + dc * dc) * (1.0f / 162.0f)); // 2*(0.15*60)^2 = 162
    L[idx] = sm * (1.0f - g);
}

// =====================================================================
// Kernel 8: per-row / per-column ops: top-10 + soft_argmax2d
//   z==0: row n of W  -> val1[b,:,n], grid1[b,n,:]
//   z==1: col m of W  -> val2[b,:,m], grid2[b,m,:]
// =====================================================================
__device__ __forceinline__ float parzenw(float d) {
    const float u = d * 0.2f;                              // d / (WIN_WIDTH/2)
    if (u <= 0.5f) return 1.0f - 6.0f * u * u * (1.0f - u);
    if (u <= 1.0f) { const float t = 1.0f - u; return 2.0f * t * t * t; }
    return 0.0f;
}

__global__ __launch_bounds__(256) void rowcol_ops(
    const float* __restrict__ W, float* __restrict__ out,
    float* __restrict__ grid1, float* __restrict__ grid2)
{
    __shared__ float row[HW];
    __shared__ float redv[256];
    __shared__ int   redi[256];
    __shared__ float su[256], sux[256], suy[256];

    const int n = blockIdx.x, b = blockIdx.y, which = blockIdx.z, t = threadIdx.x;
    const float* base;
    size_t stride;
    if (which == 0) { base = W + ((size_t)b * HW + n) * HW; stride = 1; }
    else            { base = W + (size_t)b * HW * HW + n;   stride = HW; }

    for (int e = t; e < HW; e += 256) row[e] = base[(size_t)e * stride];
    __syncthreads();

    // ---- argmax (lowest index on ties, matching jnp.argmax) ----
    float lv = -FLT_MAX; int li = 0;
    for (int e = t; e < HW; e += 256) { const float v = row[e]; if (v > lv) { lv = v; li = e; } }
    redv[t] = lv; redi[t] = li; __syncthreads();
    for (int s = 128; s > 0; s >>= 1) {
        if (t < s) {
            const float ov = redv[t + s]; const int oi = redi[t + s];
            if (ov > redv[t] || (ov == redv[t] && oi < redi[t])) { redv[t] = ov; redi[t] = oi; }
        }
        __syncthreads();
    }
    const float mx = redv[0]; const int idx = redi[0]; __syncthreads();

    // ---- softmax denominator of 5*W ----
    float ds = 0.f;
    for (int e = t; e < HW; e += 256) ds += expf(5.0f * (row[e] - mx));
    su[t] = ds; __syncthreads();
    for (int s = 128; s > 0; s >>= 1) { if (t < s) su[t] += su[t + s]; __syncthreads(); }
    const float denom = su[0]; __syncthreads();

    // ---- parzen-weighted expectation ----
    const int my = idx / 60, mcc = idx - my * 60;
    float u = 0.f, ux = 0.f, uy = 0.f;
    for (int e = t; e < HW; e += 256) {
        const int r = e / 60, c = e - r * 60;
        const float wgt = parzenw(fabsf((float)(r - my))) * parzenw(fabsf((float)(c - mcc)));
        if (wgt > 0.f) {
            const float p = expf(5.0f * (row[e] - mx)) * wgt;
            u += p; ux += p * (float)c; uy += p * (float)r;
        }
    }
    su[t] = u; sux[t] = ux; suy[t] = uy; __syncthreads();
    for (int s = 128; s > 0; s >>= 1) {
        if (t < s) { su[t] += su[t + s]; sux[t] += sux[t + s]; suy[t] += suy[t + s]; }
        __syncthreads();
    }
    if (t == 0) {
        // ex = (Ux/denom) / (U/denom + 1e-12) = Ux / (U + 1e-12*denom); grid = [ex,ey]/60
        const float P = su[0] + 1e-12f * denom;
        float* g = (which == 0 ? grid1 : grid2) + ((size_t)b * HW + n) * 2;
        g[0] = (sux[0] / P) * (1.0f / 60.0f);
        g[1] = (suy[0] / P) * (1.0f / 60.0f);
    }

    // ---- iterative top-10 with exclusion ----
    float* val = out + (which == 0 ? OFF_VAL1 : OFF_VAL2) + (size_t)b * KTOP * HW + n;
    if (t == 0) { val[0] = mx; row[idx] = -FLT_MAX; }
    __syncthreads();
    for (int k = 1; k < KTOP; ++k) {
        float lv2 = -FLT_MAX; int li2 = 0;
        for (int e = t; e < HW; e += 256) { const float v = row[e]; if (v > lv2) { lv2 = v; li2 = e; } }
        redv[t] = lv2; redi[t] = li2; __syncthreads();
        for (int s = 128; s > 0; s >>= 1) {
            if (t < s) {
                const float ov = redv[t + s]; const int oi = redi[t + s];
                if (ov > redv[t] || (ov == redv[t] && oi < redi[t])) { redv[t] = ov; redi[t] = oi; }
            }
            __syncthreads();
        }
        if (t == 0) { val[(size_t)k * HW] = redv[0]; row[redi[0]] = -FLT_MAX; }
        __syncthreads();
    }
}

// =====================================================================
// Kernel 9: channel concat + bilinear grid_sample (zeros padding)
// =====================================================================
__global__ __launch_bounds__(256) void cat_sample(
    const float* __restrict__ x1, const float* __restrict__ x2,
    const float* __restrict__ grid1, const float* __restrict__ grid2,
    float* __restrict__ out)
{
    const size_t tid = (size_t)blockIdx.x * 256 + threadIdx.x;
    const size_t total = (size_t)2 * Bb * 512 * HW;
    if (tid >= total) return;
    const int pos = (int)(tid % HW);
    size_t r = tid / HW;
    const int c = (int)(r % 512); r /= 512;
    const int b = (int)(r % Bb);
    const int which = (int)(r / Bb);

    const float* src = (which == 0) ? x1 : x2;
    const float* img = (which == 0) ? x2 : x1;
    const float* grd = (which == 0) ? grid1 : grid2;
    float* o = out + (size_t)which * Bb * 512 * HW + ((size_t)b * 512 + c) * HW + pos;

    if (c < Cc) { *o = src[((size_t)b * Cc + c) * HW + pos]; return; }

    const int ch = c - Cc;
    const float gx = grd[((size_t)b * HW + pos) * 2 + 0];
    const float gy = grd[((size_t)b * HW + pos) * 2 + 1];
    const float x = ((gx + 1.0f) * (float)Ww - 1.0f) * 0.5f;
    const float y = ((gy + 1.0f) * (float)Hh - 1.0f) * 0.5f;
    const float x0 = floorf(x), y0 = floorf(y);
    const float wx1 = x - x0, wy1 = y - y0;
    const float wx0 = 1.f - wx1, wy0 = 1.f - wy1;
    const float* ip = img + ((size_t)b * Cc + ch) * HW;

    float acc = 0.f;
    #pragma unroll
    for (int dy = 0; dy < 2; ++dy) {
        #pragma unroll
        for (int dx = 0; dx < 2; ++dx) {
            const float xf = x0 + (float)dx, yf = y0 + (float)dy;
            const bool valid = (xf >= 0.f) && (xf <= (float)(Ww - 1)) &&
                               (yf >= 0.f) && (yf <= (float)(Hh - 1));
            const int xc = (int)fminf(fmaxf(xf, 0.f), (float)(Ww - 1));
            const int yc = (int)fminf(fmaxf(yf, 0.f), (float)(Hh - 1));
            const float v = valid ? ip[yc * Ww + xc] : 0.f;
            acc += v * (dx ? wx1 : wx0) * (dy ? wy1 : wy0);
        }
    }
    *o = acc;
}

// =====================================================================
extern "C" void kernel_launch(void* const* d_in, const int* in_sizes, int n_in,
                              void* d_out, int out_size, void* d_ws, size_t ws_size,
                              hipStream_t stream)
{
    const float* x1    = (const float*)d_in[0];
    const float* x2    = (const float*)d_in[1];
    const float* alpha = (const float*)d_in[2];
    float* out = (float*)d_out;
    float* ws  = (float*)d_ws;

    // workspace layout (floats)
    float* L       = ws;                                  // Bb*HW*HW  (logits -> W in place)
    float* rowmax  = L + (size_t)Bb * HW * HW;            // Bb*HW
    float* rowsum  = rowmax + (size_t)Bb * HW;
    float* colmax  = rowsum + (size_t)Bb * HW;
    float* colsum  = colmax + (size_t)Bb * HW;
    float* grid1   = colsum + (size_t)Bb * HW;            // Bb*HW*2
    float* grid2   = grid1 + (size_t)Bb * HW * 2;         // Bb*HW*2
    float* partbuf = grid2 + (size_t)Bb * HW * 2;         // Bb*NCHUNK*HW

    // 1. logits via f32 WMMA
    gemm_wmma_f32<<<dim3(HW / 48, HW / 16, Bb), 32, 0, stream>>>(x1, x2, alpha, L);
    // 2. row softmax stats
    row_stats<<<dim3(HW, Bb), 256, 0, stream>>>(L, rowmax, rowsum);
    // 3-6. column softmax stats (deterministic two-level)
    col_max_part<<<dim3((HW + 255) / 256, NCHUNK, Bb), 256, 0, stream>>>(L, partbuf);
    col_max_comb<<<(Bb * HW + 255) / 256, 256, 0, stream>>>(partbuf, colmax);
    col_sum_part<<<dim3((HW + 255) / 256, NCHUNK, Bb), 256, 0, stream>>>(L, colmax, partbuf);
    col_sum_comb<<<(Bb * HW + 255) / 256, 256, 0, stream>>>(partbuf, colsum);
    // 7. fused dual-softmax * zero-window mask (in place)
    {
        const size_t tot = (size_t)Bb * HW * HW;
        compute_w<<<(unsigned)((tot + 255) / 256), 256, 0, stream>>>(L, rowmax, rowsum, colmax, colsum);
    }
    // 8. per-row (val1,grid1) and per-col (val2,grid2) top-10 + soft-argmax
    rowcol_ops<<<dim3(HW, Bb, 2), 256, 0, stream>>>(L, out, grid1, grid2);
    // 9. concat + bilinear sample
    {
        const size_t tot = (size_t)2 * Bb * 512 * HW;
        cat_sample<<<(unsigned)((tot + 255) / 256), 256, 0, stream>>>(x1, x2, grid1, grid2, out);
    }
}